// PreLossSampler_50070728737410
// MI455X (gfx1250) — compile-verified
//
#include <hip/hip_runtime.h>
#include <hip/hip_bf16.h>
#include <math.h>

// ---------------------------------------------------------------------------
// PreLossSampler for MI455X (gfx1250, wave32).
// Heavy cost = rotated-rect polygon clip (VALU bound). We eliminate ~97% of
// clip work with a bounding-circle distance prefilter whose pairwise center
// dot-products are computed with V_WMMA_F32_16X16X4_F32 (one 16x16 tile per
// wave, K=4 zero padded: K0=x, K1=y). The clip itself is fully register-
// resident: no dynamic array indexing, no scratch — the angle ordering is a
// selection walk of unrolled 24-way compare-selects.
// ---------------------------------------------------------------------------

typedef __attribute__((ext_vector_type(2))) float v2f;
typedef __attribute__((ext_vector_type(8))) float v8f;

#define NBOX        1024
#define REG_FG      0.55f
#define CLS_FG      0.75f
#define CLS_BG      0.25f
#define NMS_THR     0.1f
#define EPSF        1e-8f
#define PREF_MARGIN 0.1f

// ---------------------------------------------------------------------------
// Exact rotated-rectangle intersection area (faithful port of reference).
// All scalar args, all loops fully unrolled with constant indices ->
// register-resident, zero scratch. Inlined once per kernel (row loop rolled).
// ---------------------------------------------------------------------------
__device__ __forceinline__ float rect_inter_area(
    float axc, float ayc, float adx, float ady, float arr,
    float bxc, float byc, float bdx, float bdy, float brr) {
    const float sxt[4] = {1.f, 1.f, -1.f, -1.f};
    const float syt[4] = {1.f, -1.f, -1.f, 1.f};
    float acs = __cosf(arr), asn = __sinf(arr);
    float bcs = __cosf(brr), bsn = __sinf(brr);

    float acx[4], acy[4], bcx[4], bcy[4];
    {
        float hx = 0.5f * adx, hy = 0.5f * ady;
#pragma unroll
        for (int i = 0; i < 4; ++i) {
            float lx = sxt[i] * hx, ly = syt[i] * hy;
            acx[i] = lx * acs - ly * asn + axc;
            acy[i] = lx * asn + ly * acs + ayc;
        }
    }
    {
        float hx = 0.5f * bdx, hy = 0.5f * bdy;
#pragma unroll
        for (int i = 0; i < 4; ++i) {
            float lx = sxt[i] * hx, ly = syt[i] * hy;
            bcx[i] = lx * bcs - ly * bsn + bxc;
            bcy[i] = lx * bsn + ly * bcs + byc;
        }
    }

    float px[24], py[24];
    bool  val[24];
    // corners of A inside B  (rotate by -rB: lx = dx*cs + dy*sn ; ly = -dx*sn + dy*cs)
#pragma unroll
    for (int i = 0; i < 4; ++i) {
        px[i] = acx[i]; py[i] = acy[i];
        float dx = acx[i] - bxc, dy = acy[i] - byc;
        float lx =  dx * bcs + dy * bsn;
        float ly = -dx * bsn + dy * bcs;
        val[i] = (fabsf(lx) <= 0.5f * bdx + 1e-5f) && (fabsf(ly) <= 0.5f * bdy + 1e-5f);
    }
    // corners of B inside A
#pragma unroll
    for (int j = 0; j < 4; ++j) {
        px[4 + j] = bcx[j]; py[4 + j] = bcy[j];
        float dx = bcx[j] - axc, dy = bcy[j] - ayc;
        float lx =  dx * acs + dy * asn;
        float ly = -dx * asn + dy * acs;
        val[4 + j] = (fabsf(lx) <= 0.5f * adx + 1e-5f) && (fabsf(ly) <= 0.5f * ady + 1e-5f);
    }
    // 16 edge-edge intersections
#pragma unroll
    for (int i = 0; i < 4; ++i) {
        float a0x = acx[i], a0y = acy[i];
        float d1x = acx[(i + 1) & 3] - a0x, d1y = acy[(i + 1) & 3] - a0y;
#pragma unroll
        for (int j = 0; j < 4; ++j) {
            float b0x = bcx[j], b0y = bcy[j];
            float d2x = bcx[(j + 1) & 3] - b0x, d2y = bcy[(j + 1) & 3] - b0y;
            float r0x = b0x - a0x, r0y = b0y - a0y;
            float den  = d1x * d2y - d1y * d2x;
            float sden = (fabsf(den) < EPSF) ? 1.f : den;
            float rs   = 1.f / sden;                 // one reciprocal, two mults
            float t = (r0x * d2y - r0y * d2x) * rs;
            float u = (r0x * d1y - r0y * d1x) * rs;
            bool ok = (fabsf(den) > EPSF) && (t >= 0.f) && (t <= 1.f) && (u >= 0.f) && (u <= 1.f);
            int idx = 8 + i * 4 + j;
            px[idx] = a0x + t * d1x;
            py[idx] = a0y + t * d1y;
            val[idx] = ok;
        }
    }
    // centroid of valid points
    int   k = 0;
    float sxs = 0.f, sys = 0.f;
#pragma unroll
    for (int i = 0; i < 24; ++i)
        if (val[i]) { ++k; sxs += px[i]; sys += py[i]; }
    float inv = 1.f / (float)(k < 1 ? 1 : k);
    float ctx = sxs * inv, cty = sys * inv;
    float ang[24];
#pragma unroll
    for (int i = 0; i < 24; ++i) {
        float cx = val[i] ? (px[i] - ctx) : 0.f;
        float cy = val[i] ? (py[i] - cty) : 0.f;
        px[i] = cx; py[i] = cy;
        ang[i] = val[i] ? atan2f(cy, cx) : 1e9f;
    }
    // Angular selection walk: visit points in ascending (angle, index) order
    // (lexicographic tie-break == stable argsort), accumulating the shoelace
    // sum. Register-only: the 24-way scan is unrolled with constant indices.
    float curA = -1e30f; int curI = -1;
    float prevx = 0.f, prevy = 0.f, firstx = 0.f, firsty = 0.f;
    float area2 = 0.f;
    for (int s = 0; s < k; ++s) {           // k <= 24, dynamic bound
        float bA = 1e30f; int bI = 0x7fffffff;
        float bx = 0.f, by = 0.f;
#pragma unroll
        for (int i = 0; i < 24; ++i) {
            bool gt = (ang[i] > curA) || (ang[i] == curA && i > curI);
            bool lt = (ang[i] < bA)   || (ang[i] == bA   && i < bI);
            if (gt && lt) { bA = ang[i]; bI = i; bx = px[i]; by = py[i]; }
        }
        if (s == 0) { firstx = bx; firsty = by; }
        else        { area2 += prevx * by - prevy * bx; }
        prevx = bx; prevy = by;
        curA = bA; curI = bI;
    }
    area2 += prevx * firsty - prevy * firstx;   // close polygon (0 if k==0)
    float area = 0.5f * fabsf(area2);
    return (k >= 3) ? area : 0.f;
}

// ---------------------------------------------------------------------------
// Kernel 1: elementwise outputs + aux arrays (norm^2, bounding radius).
// ---------------------------------------------------------------------------
__global__ void setup_kernel(const float* __restrict__ labels,
                             const float* __restrict__ gt,    // n x 8
                             const float* __restrict__ pred,  // n x 7
                             const float* __restrict__ cls,
                             float* __restrict__ out,         // [0..n): reg_valid, [n..2n): labels
                             float2* __restrict__ auxGt,
                             float2* __restrict__ auxPred,
                             int n) {
    int i = blockIdx.x * blockDim.x + threadIdx.x;
    if (i >= n) return;
    float s   = 1.f / (1.f + __expf(-cls[i]));
    float lab = labels[i];
    out[i]     = (s > REG_FG && lab > REG_FG) ? 1.f : 0.f;
    out[n + i] = lab;
    float gx = gt[i * 8 + 0], gy = gt[i * 8 + 1], gdx = gt[i * 8 + 3], gdy = gt[i * 8 + 4];
    auxGt[i] = make_float2(gx * gx + gy * gy, 0.5f * sqrtf(gdx * gdx + gdy * gdy));
    float pxc = pred[i * 7 + 0], pyc = pred[i * 7 + 1], pdx = pred[i * 7 + 3], pdy = pred[i * 7 + 4];
    auxPred[i] = make_float2(pxc * pxc + pyc * pyc, 0.5f * sqrtf(pdx * pdx + pdy * pdy));
}

// ---------------------------------------------------------------------------
// Kernel 2: single-block bitonic argsort, descending by score, stable ties.
// ---------------------------------------------------------------------------
__global__ void __launch_bounds__(512) argsort_kernel(const float* __restrict__ scores,
                                                      int* __restrict__ order) {
    __shared__ float sk[NBOX];
    __shared__ int   si[NBOX];
    int t = threadIdx.x;
    sk[t] = scores[t];             si[t] = t;
    sk[t + 512] = scores[t + 512]; si[t + 512] = t + 512;
    __syncthreads();
    for (int k = 2; k <= NBOX; k <<= 1) {
        for (int j = k >> 1; j > 0; j >>= 1) {
            int low = t & (j - 1);
            int pos = ((t & ~(j - 1)) << 1) | low;
            int oth = pos + j;
            bool dir = ((pos & k) == 0);
            float kp = sk[pos], ko = sk[oth];
            int   ip = si[pos], io = si[oth];
            bool before_po = (kp > ko) || (kp == ko && ip < io); // pos should precede oth
            bool wrong = dir ? !before_po : before_po;
            if (wrong) { sk[pos] = ko; sk[oth] = kp; si[pos] = io; si[oth] = ip; }
            __syncthreads();
        }
    }
    order[t] = si[t];
    order[t + 512] = si[t + 512];
}

// ---------------------------------------------------------------------------
// Kernel 3/5: pairwise rotated IoU over 16x16 tiles. One wave per tile.
// WMMA f32 16x16x4 computes all 256 center dot-products; a cheap unrolled
// distance pass builds an 8-bit candidate mask; the rolled heavy loop runs
// the register-resident clip only for surviving rows.
// ---------------------------------------------------------------------------
template <bool IOU3D>
__global__ void __launch_bounds__(128) pair_iou_kernel(
    const float* __restrict__ A, int strideA, const float2* __restrict__ auxA, const int* __restrict__ ordA,
    const float* __restrict__ B, int strideB, const float2* __restrict__ auxB, const int* __restrict__ ordB,
    float* __restrict__ out, int n) {
    const int lane = threadIdx.x;           // 0..31 (wave32)
    const int ti = blockIdx.y;
    const int tj = blockIdx.x * 4 + threadIdx.y;
    const int i0 = ti * 16, j0 = tj * 16;
    const int mm = lane & 15;

    // A (16x4): lanes 0-15 hold K0=x, K1=y for M=lane; lanes 16-31 -> K2,K3 = 0.
    // B (4x16): lanes 0-15 hold K0=x, K1=y for N=lane; lanes 16-31 -> K2,K3 = 0.
    v2f av = {0.f, 0.f}, bv = {0.f, 0.f};
    if (lane < 16) {
        int ia = ordA ? ordA[i0 + mm] : (i0 + mm);
        av.x = A[ia * strideA + 0];
        av.y = A[ia * strideA + 1];
        int jb0 = ordB ? ordB[j0 + mm] : (j0 + mm);
        bv.x = B[jb0 * strideB + 0];
        bv.y = B[jb0 * strideB + 1];
    }
    v8f acc = {};
    // dot[m][n] = xa_m*xb_n + ya_m*yb_n  (full-EXEC wave32 WMMA)
    acc = __builtin_amdgcn_wmma_f32_16x16x4_f32(false, av, false, bv,
                                                (short)0, acc, false, false);

    const int j  = j0 + mm;
    const int jb = ordB ? ordB[j] : j;
    float2 xb = auxB[jb];
    float b0 = B[jb * strideB + 0], b1 = B[jb * strideB + 1];
    float b2 = B[jb * strideB + 2], b3 = B[jb * strideB + 3];
    float b4 = B[jb * strideB + 4], b5 = B[jb * strideB + 5];
    float b6 = B[jb * strideB + 6];

    const int half = lane >> 4;

    // Cheap prefilter pass (unrolled: acc[r] has constant index -> no dynamic
    // extract from the WMMA accumulator). ~2-3% of pairs survive.
    unsigned cand = 0;
#pragma unroll
    for (int r = 0; r < 8; ++r) {
        // C/D layout: VGPR r -> row r (lanes 0-15) / row r+8 (lanes 16-31), col = lane&15
        int i  = i0 + r + 8 * half;
        int ia = ordA ? ordA[i] : i;
        float2 xa = auxA[ia];
        float dsq = xa.x + xb.x - 2.f * acc[r];        // |pa - pb|^2
        float thr = xa.y + xb.y + PREF_MARGIN;          // ra + rb + margin
        if (dsq <= thr * thr) cand |= (1u << r);
    }

    // Heavy loop: rolled -> exactly one inlined copy of the clip per kernel.
    for (int r = 0; r < 8; ++r) {
        float v = 0.f;
        int i = i0 + r + 8 * half;
        if (cand & (1u << r)) {
            int ia = ordA ? ordA[i] : i;
            float a0 = A[ia * strideA + 0], a1 = A[ia * strideA + 1];
            float a2 = A[ia * strideA + 2], a3 = A[ia * strideA + 3];
            float a4 = A[ia * strideA + 4], a5 = A[ia * strideA + 5];
            float a6 = A[ia * strideA + 6];
            float inter = rect_inter_area(a0, a1, a3, a4, a6,
                                          b0, b1, b3, b4, b6);
            if (inter > 0.f) {
                if (!IOU3D) {
                    float sa = a3 * a4, sb = b3 * b4;
                    float den = sa + sb - inter;
                    den = den < EPSF ? EPSF : den;
                    v = inter / den;
                } else {
                    float amax = a2 + 0.5f * a5, amin = a2 - 0.5f * a5;
                    float bmax = b2 + 0.5f * b5, bmin = b2 - 0.5f * b5;
                    float oh = fminf(amax, bmax) - fmaxf(amin, bmin);
                    oh = oh < 0.f ? 0.f : oh;
                    float i3 = inter * oh;
                    float va = a3 * a4 * a5, vb = b3 * b4 * b5;
                    float den = va + vb - i3;
                    den = den < EPSF ? EPSF : den;
                    v = i3 / den;
                }
            }
        }
        out[(size_t)i * n + j] = v;
    }
}

// ---------------------------------------------------------------------------
// Kernel 4: greedy NMS sweep (single 1024-thread block) + scatter masked GT.
// ---------------------------------------------------------------------------
__global__ void __launch_bounds__(1024) nms_kernel(const float* __restrict__ iouO,
                                                   const int* __restrict__ order,
                                                   const float* __restrict__ gt,     // n x 8
                                                   float* __restrict__ gtm,          // n x 8
                                                   const float2* __restrict__ auxGt,
                                                   float2* __restrict__ auxGtm,
                                                   int n) {
    __shared__ int keep[NBOX];
    int t = threadIdx.x;
    keep[t] = 1;
    __syncthreads();
    for (int i = 0; i < n; ++i) {
        if (keep[i] && t > i && iouO[(size_t)i * n + t] > NMS_THR) keep[t] = 0;
        __syncthreads();
    }
    int oi = order[t];
    float m = keep[t] ? 1.f : 0.f;
#pragma unroll
    for (int c = 0; c < 8; ++c) gtm[oi * 8 + c] = m * gt[oi * 8 + c];
    float2 ax = auxGt[oi];
    auxGtm[oi] = make_float2(m * ax.x, m * ax.y);
}

// ---------------------------------------------------------------------------
// Kernel 6: per-row max/argmax (first-index tie rule) + CLS thresholds.
// ---------------------------------------------------------------------------
__global__ void __launch_bounds__(256) rowmax_kernel(const float* __restrict__ M,
                                                     float* __restrict__ out, int n) {
    int i = blockIdx.x, t = threadIdx.x;
    float best = -1.f;
    int   bidx = 0;
    for (int j = t; j < n; j += 256) {
        float v = M[(size_t)i * n + j];
        if (v > best) { best = v; bidx = j; }   // strict > keeps earliest index per thread
    }
    __shared__ float sv[256];
    __shared__ int   sidx[256];
    sv[t] = best; sidx[t] = bidx;
    __syncthreads();
    for (int s = 128; s > 0; s >>= 1) {
        if (t < s) {
            float v2 = sv[t + s];
            int   i2 = sidx[t + s];
            if (v2 > sv[t] || (v2 == sv[t] && i2 < sidx[t])) { sv[t] = v2; sidx[t] = i2; }
        }
        __syncthreads();
    }
    if (t == 0) {
        float mo = sv[0];
        mo = mo > CLS_FG ? 1.f : (mo < CLS_BG ? 0.f : mo);
        out[2 * n + i] = mo;            // max_overlaps
        out[3 * n + i] = (float)sidx[0]; // gt_assignment
    }
}

// ---------------------------------------------------------------------------
extern "C" void kernel_launch(void* const* d_in, const int* in_sizes, int n_in,
                              void* d_out, int out_size, void* d_ws, size_t ws_size,
                              hipStream_t stream) {
    const float* labels = (const float*)d_in[0];  // (N,)
    const float* pred   = (const float*)d_in[1];  // (N,7)
    const float* gt     = (const float*)d_in[2];  // (N,8)
    const float* cls    = (const float*)d_in[3];  // (N,)
    // d_in[4] (reg_valid_mask, int64) unused by the reference outputs.
    float* out = (float*)d_out;                   // 4N floats: reg_valid | labels | max_ov | gt_assign
    const int n = in_sizes[0];                    // 1024

    // workspace layout (8-byte aligned slices)
    char* ws = (char*)d_ws;
    int*    order   = (int*)(ws + 0);                       // 4 KB
    float2* auxGt   = (float2*)(ws + 4096);                 // 8 KB
    float2* auxPred = (float2*)(ws + 12288);                // 8 KB
    float2* auxGtm  = (float2*)(ws + 20480);                // 8 KB
    float*  gtm     = (float*)(ws + 28672);                 // 32 KB (n x 8)
    float*  iouO    = (float*)(ws + 61440);                 // 4 MB
    float*  iou3    = (float*)(ws + 61440 + (size_t)n * n * 4); // 4 MB

    // 1) elementwise outputs + aux
    setup_kernel<<<(n + 255) / 256, 256, 0, stream>>>(labels, gt, pred, cls,
                                                      out, auxGt, auxPred, n);
    // 2) argsort scores descending
    argsort_kernel<<<1, 512, 0, stream>>>(labels, order);
    // 3) BEV IoU of ordered GT x ordered GT (WMMA-prefiltered tiles)
    {
        dim3 grid(n / 64, n / 16), block(32, 4);
        pair_iou_kernel<false><<<grid, block, 0, stream>>>(
            gt, 8, auxGt, order, gt, 8, auxGt, order, iouO, n);
    }
    // 4) greedy NMS + build masked GT
    nms_kernel<<<1, 1024, 0, stream>>>(iouO, order, gt, gtm, auxGt, auxGtm, n);
    // 5) 3D IoU of pred x masked GT (WMMA-prefiltered tiles)
    {
        dim3 grid(n / 64, n / 16), block(32, 4);
        pair_iou_kernel<true><<<grid, block, 0, stream>>>(
            pred, 7, auxPred, nullptr, gtm, 8, auxGtm, nullptr, iou3, n);
    }
    // 6) row max / argmax + thresholds
    rowmax_kernel<<<n, 256, 0, stream>>>(iou3, out, n);
}